// AttentionLayer_4200478016044
// MI455X (gfx1250) — compile-verified
//
#include <hip/hip_runtime.h>
#include <hip/hip_bf16.h>
#include <math.h>

// Problem constants (match reference)
#define NSTEP 64
#define H2    1024
#define TT    512
#define EPSN  1e-12f

typedef __attribute__((ext_vector_type(2))) float v2f;
typedef __attribute__((ext_vector_type(8))) float v8f;

// ---------------------------------------------------------------------------
// Kernel A: sequential recurrence  u <- normalize(W @ u + b), 64 steps.
// One workgroup (1024 threads = 32 waves). W rows read coalesced: wave w owns
// rows e = w*32..w*32+31; lanes split the K dimension.
// ---------------------------------------------------------------------------
__global__ void __launch_bounds__(1024)
recurrence_kernel(const float* __restrict__ W, const float* __restrict__ bias,
                  const float* __restrict__ u0, float* __restrict__ u_ws)
{
    __shared__ float u[H2];
    __shared__ float y[H2];
    __shared__ float red[32];

    const int tid  = threadIdx.x;
    const int wave = tid >> 5;
    const int lane = tid & 31;

    u[tid] = u0[tid];
    __syncthreads();

    for (int i = 0; i < NSTEP; ++i) {
        // y[e] = W[e,:] . u + b[e], 32 rows per wave, coalesced over lanes
        for (int er = 0; er < 32; ++er) {
            const int e = wave * 32 + er;
            const float* wr = W + (size_t)e * H2;
            float p = 0.0f;
            #pragma unroll 4
            for (int d = lane; d < H2; d += 32)
                p += wr[d] * u[d];
            #pragma unroll
            for (int off = 16; off >= 1; off >>= 1)
                p += __shfl_xor(p, off, 32);
            if (lane == 0) y[e] = p + bias[e];
        }
        __syncthreads();

        // global norm over y
        float v  = y[tid];
        float sq = v * v;
        #pragma unroll
        for (int off = 16; off >= 1; off >>= 1)
            sq += __shfl_xor(sq, off, 32);
        if (lane == 0) red[wave] = sq;
        __syncthreads();
        if (tid < 32) {
            float r = red[tid];
            #pragma unroll
            for (int off = 16; off >= 1; off >>= 1)
                r += __shfl_xor(r, off, 32);
            red[tid] = r;
        }
        __syncthreads();

        const float inv = 1.0f / fmaxf(sqrtf(red[0]), EPSN);
        const float val = v * inv;
        u_ws[(size_t)i * H2 + tid] = val;
        u[tid] = val;
        __syncthreads();
    }
}

// ---------------------------------------------------------------------------
// Kernel B: fused  scores[n,t] = sum_e tanh( (h[n,:,t] . W[e,:]) + b[e] ) * u_ws[n,e]
// One block per (n, 16-token tile). 8 waves; each wave owns 8 e-tiles of 16.
// A-strip of hidden staged in LDS (K-chunked), fp32 WMMA 16x16x4 over K=1024.
// u_it is never materialized.
// ---------------------------------------------------------------------------
#define KC   512           // K chunk staged in LDS
#define KPAD (KC + 2)      // pad rows: lane banks spaced by 2 dwords

__global__ void __launch_bounds__(256)
scores_kernel(const float* __restrict__ hidden, const float* __restrict__ W,
              const float* __restrict__ bias,  const float* __restrict__ u_ws,
              float* __restrict__ scores)
{
    __shared__ float sA[16 * KPAD];   // ~33 KB: sA[t][k_local]
    __shared__ float sScore[16];

    const int n    = blockIdx.x;
    const int t0   = blockIdx.y * 16;
    const int tid  = threadIdx.x;
    const int lane = tid & 31;
    const int wave = tid >> 5;              // 0..7
    const int l16  = lane & 15;             // M (=t) for A, N (=e) for B/C/D
    const int kHalf = (lane >> 4) * 2;      // K sub-offset: 0 (lanes 0-15) / 2 (16-31)

    if (tid < 16) sScore[tid] = 0.0f;

    const float* hbase = hidden + (size_t)n * H2 * TT + t0;

    // 8 accumulator tiles per wave: e-tile et = wave + idx*8
    v8f accs[8];
    #pragma unroll
    for (int i = 0; i < 8; ++i)
        accs[i] = (v8f){0.f,0.f,0.f,0.f,0.f,0.f,0.f,0.f};

    for (int kc = 0; kc < H2; kc += KC) {
        // stage hidden[n, kc:kc+KC, t0:t0+16] as sA[t][k]
        for (int i = tid; i < KC * 16; i += 256) {
            const int k = i >> 4;
            const int t = i & 15;
            sA[t * KPAD + k] = hbase[(size_t)(kc + k) * TT + t];
        }
        __syncthreads();

        #pragma unroll
        for (int idx = 0; idx < 8; ++idx) {
            const int e0 = (wave + idx * 8) * 16;
            const float* Wrow = W + (size_t)(e0 + l16) * H2 + kc + kHalf;
            const float* Arow = sA + l16 * KPAD + kHalf;
            v8f acc = accs[idx];
            for (int k0 = 0; k0 < KC; k0 += 4) {
                v2f av = *(const v2f*)(Arow + k0);   // A: 16x4 fp32 fragment
                v2f bv = *(const v2f*)(Wrow + k0);   // B: 4x16 fp32 fragment (W^T tile)
                acc = __builtin_amdgcn_wmma_f32_16x16x4_f32(
                        /*neg_a=*/false, av, /*neg_b=*/false, bv,
                        /*c_mod=*/(short)0, acc,
                        /*reuse_a=*/false, /*reuse_b=*/false);
            }
            accs[idx] = acc;
        }
        __syncthreads();
    }

    // epilogue: tanh(+bias) * u_ws, reduce over e
    float tAcc[8];
    #pragma unroll
    for (int r = 0; r < 8; ++r) tAcc[r] = 0.0f;

    #pragma unroll
    for (int idx = 0; idx < 8; ++idx) {
        const int e0 = (wave + idx * 8) * 16;
        const float bb = bias[e0 + l16];
        const float uu = u_ws[(size_t)n * H2 + e0 + l16];
        #pragma unroll
        for (int r = 0; r < 8; ++r)
            tAcc[r] += tanhf(accs[idx][r] + bb) * uu;
    }

    // lane-reduce over the 16 N-lanes of each half (t=r in lanes 0-15, t=r+8 in 16-31)
    #pragma unroll
    for (int r = 0; r < 8; ++r) {
        float v = tAcc[r];
        #pragma unroll
        for (int off = 8; off >= 1; off >>= 1)
            v += __shfl_xor(v, off, 32);
        if (l16 == 0) {
            const int t = r + ((lane >> 4) << 3);
            atomicAdd(&sScore[t], v);          // ds_add_f32
        }
    }
    __syncthreads();

    if (tid < 16)
        scores[(size_t)n * TT + t0 + tid] = sScore[tid];
}

// ---------------------------------------------------------------------------
// Kernel C: softmax over T per step n. One block (512 threads = 16 waves) per n.
// ---------------------------------------------------------------------------
__global__ void __launch_bounds__(512)
softmax_kernel(const float* __restrict__ scores, float* __restrict__ attn)
{
    __shared__ float red[16];
    const int n   = blockIdx.x;
    const int tid = threadIdx.x;

    const float s = scores[(size_t)n * TT + tid];

    float m = s;
    #pragma unroll
    for (int off = 16; off >= 1; off >>= 1)
        m = fmaxf(m, __shfl_xor(m, off, 32));
    if ((tid & 31) == 0) red[tid >> 5] = m;
    __syncthreads();
    if (tid < 16) {
        float r = red[tid];
        #pragma unroll
        for (int off = 8; off >= 1; off >>= 1)
            r = fmaxf(r, __shfl_xor(r, off, 32));
        red[tid] = r;
    }
    __syncthreads();
    m = red[0];
    __syncthreads();

    const float e = __expf(s - m);
    float sum = e;
    #pragma unroll
    for (int off = 16; off >= 1; off >>= 1)
        sum += __shfl_xor(sum, off, 32);
    if ((tid & 31) == 0) red[tid >> 5] = sum;
    __syncthreads();
    if (tid < 16) {
        float r = red[tid];
        #pragma unroll
        for (int off = 8; off >= 1; off >>= 1)
            r += __shfl_xor(r, off, 32);
        red[tid] = r;
    }
    __syncthreads();

    attn[(size_t)n * TT + tid] = e / red[0];
}

// ---------------------------------------------------------------------------
// Kernel D: s[n,d] = sum_t hidden[n,d,t] * a[n,t]. Attention row staged in LDS,
// float4 streaming over contiguous T.
// ---------------------------------------------------------------------------
__global__ void __launch_bounds__(256)
weighted_sum_kernel(const float* __restrict__ hidden,
                    const float* __restrict__ attn, float* __restrict__ out)
{
    __shared__ float sa[TT];
    const int n   = blockIdx.x;
    const int dc  = blockIdx.y;
    const int tid = threadIdx.x;

    sa[tid]       = attn[(size_t)n * TT + tid];
    sa[tid + 256] = attn[(size_t)n * TT + tid + 256];
    __syncthreads();

    const int d = dc * 256 + tid;
    const float* h = hidden + (size_t)n * H2 * TT + (size_t)d * TT;
    float acc = 0.0f;
    #pragma unroll 4
    for (int t = 0; t < TT; t += 4) {
        const float4 h4 = *(const float4*)(h + t);
        acc += h4.x * sa[t] + h4.y * sa[t + 1] + h4.z * sa[t + 2] + h4.w * sa[t + 3];
    }
    out[(size_t)n * H2 + d] = acc;
}

// ---------------------------------------------------------------------------
extern "C" void kernel_launch(void* const* d_in, const int* in_sizes, int n_in,
                              void* d_out, int out_size, void* d_ws, size_t ws_size,
                              hipStream_t stream)
{
    const float* hidden = (const float*)d_in[0];  // (N, H2, T)
    const float* W      = (const float*)d_in[1];  // (H2, H2)
    const float* bias   = (const float*)d_in[2];  // (H2,)
    const float* u0     = (const float*)d_in[3];  // (H2,)
    float* out = (float*)d_out;                   // (N, H2)

    float* ws     = (float*)d_ws;
    float* u_ws   = ws;                           // N*H2
    float* scores = u_ws + (size_t)NSTEP * H2;    // N*T
    float* attn   = scores + (size_t)NSTEP * TT;  // N*T

    recurrence_kernel<<<1, 1024, 0, stream>>>(W, bias, u0, u_ws);

    scores_kernel<<<dim3(NSTEP, TT / 16), 256, 0, stream>>>(
        hidden, W, bias, u_ws, scores);

    softmax_kernel<<<NSTEP, TT, 0, stream>>>(scores, attn);

    weighted_sum_kernel<<<dim3(NSTEP, H2 / 256), 256, 0, stream>>>(
        hidden, attn, out);
}